// ScaleDotProductAttention_70695161692391
// MI455X (gfx1250) — compile-verified
//
#include <hip/hip_runtime.h>
#include <math.h>

typedef __attribute__((ext_vector_type(2))) float v2f;
typedef __attribute__((ext_vector_type(8))) float v8f;

#define S_LEN   2048
#define DHEAD   64
#define QTILE   16
#define ST      2052                 // padded LDS row stride (floats): 2052%64==4 -> no bank conflicts
#define LDS_RED   (16 * ST)          // 32832: reduction scratch [16][8]
#define LDS_RINV  (LDS_RED + 128)    // 32960: 1/rowsum per row
#define LDS_FLOATS (LDS_RINV + 16)   // 32976 floats = 131904 bytes

__device__ __forceinline__ v8f wmma_f32(v2f a, v2f b, v8f c) {
  // D = A(16x4,f32) * B(4x16,f32) + C(16x16,f32)
  return __builtin_amdgcn_wmma_f32_16x16x4_f32(false, a, false, b, (short)0, c,
                                               false, false);
}

__launch_bounds__(128)
__global__ void sdpa_fp32_wmma(const float* __restrict__ Q,
                               const float* __restrict__ K,
                               const float* __restrict__ V,
                               float* __restrict__ Octx,
                               float* __restrict__ Oattn) {
  extern __shared__ __align__(16) float lds[];
  float* Sl   = lds;               // [16][ST] score / P tile
  float* red  = lds + LDS_RED;     // [16*8]
  float* rinv = lds + LDS_RINV;    // [16]

  const int tid   = threadIdx.x;
  const int lane  = tid & 31;
  const int wv    = tid >> 5;                 // wave 0..3
  const int h     = blockIdx.x >> 7;          // / (2048/16)
  const int q0    = (blockIdx.x & 127) << 4;  // query tile start

  const int mrow  = lane & 15;                // A-frag M / B-frag N / C-frag N
  const int khalf = (lane < 16) ? 0 : 2;      // K-pair select per lane half
  const int rb    = (lane < 16) ? 0 : 8;      // C/D row base

  const float* Qg = Q + ((size_t)h * S_LEN + q0) * DHEAD;
  const float* Kg = K + (size_t)h * S_LEN * DHEAD;
  const float* Vg = V + (size_t)h * S_LEN * DHEAD;

  // ---- preload Q A-fragments: 16 k-steps cover D=64 (32 VGPRs) ----
  v2f qa[16];
  #pragma unroll
  for (int s = 0; s < 16; ++s)
    qa[s] = *(const v2f*)(Qg + mrow * DHEAD + s * 4 + khalf);

  // ---- Phase 1: S = (Q K^T) * 1/sqrt(D) into LDS; each wave owns 32 n-blocks ----
  const float scale = 0.125f;
  for (int p = 0; p < 16; ++p) {
    const int n0a = (wv * 32 + 2 * p) * 16;
    const int n0b = n0a + 16;
    v8f acc0 = {}, acc1 = {};
    #pragma unroll
    for (int s = 0; s < 16; ++s) {
      const int kb = s * 4 + khalf;
      v2f b0 = *(const v2f*)(Kg + (size_t)(n0a + mrow) * DHEAD + kb);
      v2f b1 = *(const v2f*)(Kg + (size_t)(n0b + mrow) * DHEAD + kb);
      acc0 = wmma_f32(qa[s], b0, acc0);
      acc1 = wmma_f32(qa[s], b1, acc1);
    }
    #pragma unroll
    for (int r = 0; r < 8; ++r) {
      Sl[(rb + r) * ST + n0a + mrow] = acc0[r] * scale;
      Sl[(rb + r) * ST + n0b + mrow] = acc1[r] * scale;
    }
  }
  __syncthreads();

  // ---- Phase 2: row softmax (128 threads: 8 threads x 256 cols per row) ----
  {
    const int r  = tid & 15;
    const int j  = tid >> 4;
    const int c0 = j * 256;

    float m = -INFINITY;
    for (int c = 0; c < 256; ++c) m = fmaxf(m, Sl[r * ST + c0 + c]);
    red[r * 8 + j] = m;
    __syncthreads();
    float rm = red[r * 8];
    #pragma unroll
    for (int i = 1; i < 8; ++i) rm = fmaxf(rm, red[r * 8 + i]);
    __syncthreads();                       // done reading maxima before reuse

    float ps = 0.f;
    for (int c = 0; c < 256; ++c) {
      float e = __expf(Sl[r * ST + c0 + c] - rm);
      Sl[r * ST + c0 + c] = e;
      ps += e;
    }
    red[r * 8 + j] = ps;
    __syncthreads();
    float rs = 0.f;
    #pragma unroll
    for (int i = 0; i < 8; ++i) rs += red[r * 8 + i];
    if (j == 0) rinv[r] = 1.0f / rs;
    __syncthreads();

    // normalize + single coalesced global write of attn (128B per wave-store)
    float* attn_tile = Oattn + ((size_t)h * S_LEN + q0) * S_LEN;
    for (int i = 0; i < 256; ++i) {
      const int idx = tid + i * 128;
      const int row = idx >> 11;
      const int col = idx & 2047;
      float pv = Sl[row * ST + col] * rinv[row];
      Sl[row * ST + col] = pv;
      attn_tile[(size_t)row * S_LEN + col] = pv;
    }
  }
  __syncthreads();

  // ---- Phase 3: context = P V ; wave wv owns D-columns [wv*16, wv*16+16) ----
  {
    const int colv = wv * 16 + mrow;
    v8f acc[4] = {};
    for (int step4 = 0; step4 < 512; step4 += 4) {
      #pragma unroll
      for (int u = 0; u < 4; ++u) {
        const int kb = (step4 + u) * 4 + khalf;
        v2f a = *(const v2f*)(Sl + mrow * ST + kb);   // padded stride: conflict-free
        v2f b;
        b.x = Vg[(size_t)kb * DHEAD + colv];
        b.y = Vg[(size_t)(kb + 1) * DHEAD + colv];
        acc[u] = wmma_f32(a, b, acc[u]);
      }
    }
    v8f res = acc[0] + acc[1] + acc[2] + acc[3];
    float* Og = Octx + ((size_t)h * S_LEN + q0) * DHEAD;
    #pragma unroll
    for (int r = 0; r < 8; ++r)
      Og[(size_t)(rb + r) * DHEAD + colv] = res[r];
  }
}

extern "C" void kernel_launch(void* const* d_in, const int* in_sizes, int n_in,
                              void* d_out, int out_size, void* d_ws, size_t ws_size,
                              hipStream_t stream) {
  const float* Q = (const float*)d_in[0];
  const float* K = (const float*)d_in[1];
  const float* V = (const float*)d_in[2];
  float* ctx  = (float*)d_out;
  float* attn = ctx + (size_t)16 * 2048 * 64;   // context first, then attn

  const size_t ldsBytes = (size_t)LDS_FLOATS * sizeof(float);  // 131904 B
  hipFuncSetAttribute(reinterpret_cast<const void*>(sdpa_fp32_wmma),
                      hipFuncAttributeMaxDynamicSharedMemorySize,
                      (int)ldsBytes);

  dim3 grid(16 * (2048 / 16));   // 2048 workgroups: (head, q-tile)
  dim3 block(128);               // 4 waves
  sdpa_fp32_wmma<<<grid, block, ldsBytes, stream>>>(Q, K, V, ctx, attn);
}